// BatchHighOrderActivationB_16741782520155
// MI455X (gfx1250) — compile-verified
//
#include <hip/hip_runtime.h>
#include <stdint.h>

#ifndef __has_builtin
#define __has_builtin(x) 0
#endif

// Problem shape (from reference)
#define ARITY  3
#define GROUPS 256
#define OUT_D  8
#define BATCHN 16384
#define PROWS  27                    // 3^ARITY

// Padded LDS layout: row stride 12 floats (48B) so ds_load_b128 superbank
// = (gl*1 + idx*3) mod 16 -> full superbank coverage, ~2-way worst conflicts.
#define RSTRIDE 12                   // floats per row in LDS (8 data + 4 pad)
#define GSTRIDE (PROWS * RSTRIDE)    // 324 floats per group in LDS

// Tiling
#define GPB   64                     // groups per block
#define BPB   64                     // batch rows per block
#define NTHR  256                    // 8 wave32
#define LDSF  (GPB * GSTRIDE)        // 20736 floats = 82,944 B

typedef int   v4i __attribute__((vector_size(16)));
typedef float v4f __attribute__((ext_vector_type(4)));

// ---- CDNA5 async global->LDS copy (ASYNCcnt path) ----------------------
__device__ __forceinline__ void async_ld_b128(const void* gsrc, void* ldst) {
#if __has_builtin(__builtin_amdgcn_global_load_async_to_lds_b128)
  __builtin_amdgcn_global_load_async_to_lds_b128(
      (v4i*)gsrc, (v4i*)ldst, /*offset=*/0, /*cpol=*/0);
#else
  unsigned int       lofs = (unsigned int)(uintptr_t)ldst;
  unsigned long long ga   = (unsigned long long)(uintptr_t)gsrc;
  asm volatile("global_load_async_to_lds_b128 %0, %1, off"
               :: "v"(lofs), "v"(ga) : "memory");
#endif
}

__device__ __forceinline__ void wait_async0() {
#if __has_builtin(__builtin_amdgcn_s_wait_asynccnt)
  __builtin_amdgcn_s_wait_asynccnt(0);
#else
  asm volatile("s_wait_asynccnt 0" ::: "memory");
#endif
}

// Branchless stable compare-exchange on (|a|, signed-ternary-term) pairs
__device__ __forceinline__ void cswap(float& ka, int& ta, float& kb, int& tb) {
  bool  sw   = ka > kb;
  float kmin = sw ? kb : ka;
  float kmax = sw ? ka : kb;
  int   tmin = sw ? tb : ta;
  int   tmax = sw ? ta : tb;
  ka = kmin; kb = kmax; ta = tmin; tb = tmax;
}

__global__ __launch_bounds__(NTHR)
void hoact_kernel(const float* __restrict__ X,
                  const float* __restrict__ P,
                  float* __restrict__ Y) {
  __shared__ float sP[LDSF];

  const int tid  = threadIdx.x;
  const int bblk = blockIdx.x;   // batch tile: 0..255
  const int gblk = blockIdx.y;   // group tile: 0..3

  // Stage this block's 64-group params slice into LDS via the CDNA5 async
  // DMA path. Global side: contiguous 16B chunks (perfectly coalesced).
  // LDS side: scattered into the padded (48B-row-stride) layout; every
  // destination stays 16B-aligned.
  {
    const char* src = (const char*)(P + (size_t)gblk * (GPB * PROWS * OUT_D));
    const int NCHUNK = GPB * PROWS * 2;              // 3456 x 16B
    for (int k = tid; k < NCHUNK; k += NTHR) {
      const int row  = k >> 1;                       // global row index
      const int grp  = row / PROWS;
      const int r    = row - grp * PROWS;
      const int dstd = grp * GSTRIDE + r * RSTRIDE + (k & 1) * 4;  // dwords
      async_ld_b128(src + (size_t)k * 16, (char*)sP + (size_t)dstd * 4);
    }
    wait_async0();
  }
  __syncthreads();

  const int gl = tid & (GPB - 1);          // group within tile
  const int bl = tid >> 6;                 // 0..3 batch sub-row
  const int g  = gblk * GPB + gl;
  const float* __restrict__ gp = &sP[gl * GSTRIDE];

  for (int it = 0; it < BPB; it += NTHR / GPB) {
    const int b = bblk * BPB + it + bl;

    // 3 scalar loads; lane-stride 12B => contiguous cachelines per wave.
    // Non-temporal: X is streamed once, keep L2 for params.
    const float* xp = X + (size_t)b * (GROUPS * ARITY) + g * ARITY;
    const float a0 = __builtin_nontemporal_load(xp + 0);
    const float a1 = __builtin_nontemporal_load(xp + 1);
    const float a2 = __builtin_nontemporal_load(xp + 2);

    float k0 = fabsf(a0), k1 = fabsf(a1), k2 = fabsf(a2);
    int   t0 = (a0 >= 0.0f) ? 1 : -1;      // sign * 3^0
    int   t1 = (a1 >= 0.0f) ? 3 : -3;      // sign * 3^1
    int   t2 = (a2 >= 0.0f) ? 9 : -9;      // sign * 3^2
    cswap(k0, t0, k1, t1);
    cswap(k1, t1, k2, t2);
    cswap(k0, t0, k1, t1);                 // ascending by |a|, stable

    // idx[k] = suffix_sum(terms)[k] + 13, all in [0, 26]
    const int i2 = t2 + 13;
    const int i1 = i2 + t1;
    const int i0 = i1 + t0;
    const float c0 = k0, c1 = k1 - k0, c2 = k2 - k1;

    // Padded 48B LDS rows -> ds_load_b128 x2 per gathered row
    const v4f* r0 = (const v4f*)(gp + i0 * RSTRIDE);
    const v4f* r1 = (const v4f*)(gp + i1 * RSTRIDE);
    const v4f* r2 = (const v4f*)(gp + i2 * RSTRIDE);
    const v4f q0a = r0[0], q0b = r0[1];
    const v4f q1a = r1[0], q1b = r1[1];
    const v4f q2a = r2[0], q2b = r2[1];

    v4f oa, ob;
    oa.x = fmaf(c2, q2a.x, fmaf(c1, q1a.x, c0 * q0a.x));
    oa.y = fmaf(c2, q2a.y, fmaf(c1, q1a.y, c0 * q0a.y));
    oa.z = fmaf(c2, q2a.z, fmaf(c1, q1a.z, c0 * q0a.z));
    oa.w = fmaf(c2, q2a.w, fmaf(c1, q1a.w, c0 * q0a.w));
    ob.x = fmaf(c2, q2b.x, fmaf(c1, q1b.x, c0 * q0b.x));
    ob.y = fmaf(c2, q2b.y, fmaf(c1, q1b.y, c0 * q0b.y));
    ob.z = fmaf(c2, q2b.z, fmaf(c1, q1b.z, c0 * q0b.z));
    ob.w = fmaf(c2, q2b.w, fmaf(c1, q1b.w, c0 * q0b.w));

    // 32B-aligned output -> 2x global_store_b128 (non-temporal: write-once
    // stream, don't pollute L2), contiguous across lanes.
    v4f* op = (v4f*)(Y + (size_t)b * (GROUPS * OUT_D) + g * OUT_D);
    __builtin_nontemporal_store(oa, op + 0);
    __builtin_nontemporal_store(ob, op + 1);
  }
}

extern "C" void kernel_launch(void* const* d_in, const int* in_sizes, int n_in,
                              void* d_out, int out_size, void* d_ws, size_t ws_size,
                              hipStream_t stream) {
  (void)in_sizes; (void)n_in; (void)d_ws; (void)ws_size; (void)out_size;
  const float* X = (const float*)d_in[0];   // (16384, 768) f32
  const float* P = (const float*)d_in[1];   // (256, 27, 8) f32
  float*       Y = (float*)d_out;           // (16384, 2048) f32

  dim3 grid(BATCHN / BPB, GROUPS / GPB);    // (256, 4)
  hoact_kernel<<<grid, NTHR, 0, stream>>>(X, P, Y);
}